// GCN_44719199486541
// MI455X (gfx1250) — compile-verified
//
#include <hip/hip_runtime.h>
#include <hip/hip_bf16.h>

// ---------------------------------------------------------------------------
// GCN (2-layer) for MI455X / gfx1250.
//  - GEMMs use V_WMMA_F32_16X16X4_F32 (full-precision fp32 tensor op).
//    Shapes are compile-time (template) so all fragment addressing folds
//    into immediate offsets on global_load -> minimal VALU/SALU in the loop.
//  - Edge aggregation is L2-resident on this chip (h = 25.6MB << 192MB L2),
//    done with coalesced float4 gathers + global f32 atomics.
// ---------------------------------------------------------------------------

typedef __attribute__((ext_vector_type(2))) float v2f;
typedef __attribute__((ext_vector_type(8))) float v8f;

#define N_NODES 50000   // divisible by 16 (3125 row tiles)
#define F_INF   256
#define HIDF    128
#define NCLS    16

#if __has_builtin(__builtin_amdgcn_wmma_f32_16x16x4_f32)
#define HAVE_WMMA_F32 1
#else
#define HAVE_WMMA_F32 0
#endif

// ---------------- degree / normalization ----------------------------------

__global__ void gcn_deg_count(const int* __restrict__ dst, int E, int* __restrict__ cnt) {
    int i = blockIdx.x * blockDim.x + threadIdx.x;
    if (i < E) atomicAdd(&cnt[dst[i]], 1);
}

__global__ void gcn_dinv(const int* __restrict__ cnt, float* __restrict__ dinv, int n) {
    int i = blockIdx.x * blockDim.x + threadIdx.x;
    if (i < n) dinv[i] = rsqrtf((float)(cnt[i] + 1));   // +1 = self loop; always > 0
}

// ---------------- fp32 WMMA GEMM: C[M x NCOLS] = A[M x K] * B[K x NCOLS] ----
// One wave per 16x16 output tile. M % 16 == 0, NCOLS % 16 == 0, K % 32 == 0.
// K and NCOLS are compile-time so every load inside the unrolled chunk uses
// an immediate byte offset from a single advancing base pointer.
// EXEC is all ones (no divergence) as the WMMA ISA requires.

template <int K, int NCOLS>
__global__ void gcn_wmma_gemm_f32(const float* __restrict__ A,
                                  const float* __restrict__ B,
                                  float* __restrict__ C) {
    const int rowBase = blockIdx.x * 16;
    const int colBase = blockIdx.y * 16;
    const int lane = threadIdx.x;      // 0..31
    const int half = lane >> 4;        // 0: lanes 0-15, 1: lanes 16-31
    const int l16  = lane & 15;

    v8f acc = {0.f, 0.f, 0.f, 0.f, 0.f, 0.f, 0.f, 0.f};

#if HAVE_WMMA_F32
    // A fragment pointer for this lane: row = rowBase+l16, k offset half*2
    const float* arow = A + (size_t)(rowBase + l16) * K + half * 2;
    // B fragment pointer: column colBase+l16, row offset half*2
    const float* bcol = B + (size_t)(half * 2) * NCOLS + colBase + l16;

    // 8 WMMA per chunk; all offsets below are compile-time immediates.
#pragma unroll 1
    for (int k = 0; k < K; k += 32) {
#pragma unroll
        for (int kk = 0; kk < 32; kk += 4) {
            // A 16x4: lane<16 -> A[row][k+kk], A[row][k+kk+1]
            //         lane>=16 -> A[row][k+kk+2], A[row][k+kk+3]
            v2f a = *(const v2f*)(arow + kk);
            // B 4x16: lane<16 -> B[k+kk][col], B[k+kk+1][col]
            //         lane>=16 -> B[k+kk+2][col], B[k+kk+3][col]
            v2f b;
            b.x = bcol[(size_t)(kk + 0) * NCOLS];
            b.y = bcol[(size_t)(kk + 1) * NCOLS];
            acc = __builtin_amdgcn_wmma_f32_16x16x4_f32(
                /*neg_a=*/false, a, /*neg_b=*/false, b,
                /*c_mod=*/(short)0, acc, /*reuse_a=*/false, /*reuse_b=*/false);
        }
        arow += 32;
        bcol += (size_t)32 * NCOLS;
    }
#else
    // Scalar fallback with identical C layout (lane l16 = column, rows r/half*8)
    for (int k = 0; k < K; k += 4) {
#pragma unroll
        for (int r = 0; r < 8; ++r) {
            int row = rowBase + r + half * 8;
#pragma unroll
            for (int kk = 0; kk < 4; ++kk)
                acc[r] += A[(size_t)row * K + k + kk] * B[(size_t)(k + kk) * NCOLS + colBase + l16];
        }
    }
#endif

    // C/D layout: VGPR r -> row r (lanes 0-15) / row r+8 (lanes 16-31), col = l16
    float* crow = C + (size_t)(rowBase + half * 8) * NCOLS + colBase + l16;
#pragma unroll
    for (int r = 0; r < 8; ++r)
        crow[(size_t)r * NCOLS] = acc[r];
}

// ---------------- layer-1 edge scatter: agg[d] += h[s] * norm --------------
// One wave per edge: 32 lanes x float4 = 128 features. Gather is one
// contiguous 512B row (L2-resident); updates are f32 atomics into L2.

__global__ void gcn_agg_hid(const int* __restrict__ src, const int* __restrict__ dst,
                            const float* __restrict__ dinv,
                            const float* __restrict__ h,
                            float* __restrict__ agg, int E) {
    int e = blockIdx.x * (blockDim.x >> 5) + (threadIdx.x >> 5);
    if (e >= E) return;
    int lane = threadIdx.x & 31;
    int s = src[e], d = dst[e];
    float norm = dinv[s] * dinv[d];
    const float4* hv = (const float4*)(h + (size_t)s * HIDF);
    float* av = agg + (size_t)d * HIDF;
    float4 v = hv[lane];
    int base = lane * 4;
    atomicAdd(&av[base + 0], v.x * norm);
    atomicAdd(&av[base + 1], v.y * norm);
    atomicAdd(&av[base + 2], v.z * norm);
    atomicAdd(&av[base + 3], v.w * norm);
}

// ---------------- layer-1 finalize: self-loop + bias + ReLU (in place) -----

__global__ void gcn_final1(const float* __restrict__ h, const float* __restrict__ dinv,
                           const float* __restrict__ b1, float* __restrict__ agg) {
    int idx = blockIdx.x * blockDim.x + threadIdx.x;    // N*128 = 6.4M
    if (idx >= N_NODES * HIDF) return;
    int i = idx >> 7;
    int f = idx & (HIDF - 1);
    float di = dinv[i];
    float v = agg[idx] + h[idx] * di * di + b1[f];
    agg[idx] = v > 0.f ? v : 0.f;
}

// ---------------- layer-2 edge scatter (16 features) -----------------------

__global__ void gcn_agg_out(const int* __restrict__ src, const int* __restrict__ dst,
                            const float* __restrict__ dinv,
                            const float* __restrict__ h2,
                            float* __restrict__ agg2, int E) {
    int idx = blockIdx.x * blockDim.x + threadIdx.x;    // E*16 = 25.6M
    if (idx >= E * NCLS) return;
    int e = idx >> 4;
    int f = idx & (NCLS - 1);
    int s = src[e], d = dst[e];
    float norm = dinv[s] * dinv[d];
    atomicAdd(&agg2[(size_t)d * NCLS + f], h2[(size_t)s * NCLS + f] * norm);
}

// ---------------- layer-2 finalize into d_out ------------------------------

__global__ void gcn_final2(const float* __restrict__ h2, const float* __restrict__ dinv,
                           const float* __restrict__ b2,
                           const float* __restrict__ agg2, float* __restrict__ out) {
    int idx = blockIdx.x * blockDim.x + threadIdx.x;    // N*16 = 800000
    if (idx >= N_NODES * NCLS) return;
    int i = idx >> 4;
    int f = idx & (NCLS - 1);
    float di = dinv[i];
    out[idx] = agg2[idx] + h2[idx] * di * di + b2[f];
}

// ---------------------------------------------------------------------------

extern "C" void kernel_launch(void* const* d_in, const int* in_sizes, int n_in,
                              void* d_out, int out_size, void* d_ws, size_t ws_size,
                              hipStream_t stream) {
    const float* x  = (const float*)d_in[0];   // [N, 256]
    const int*   ei = (const int*)d_in[1];     // [2, E]
    const float* W1 = (const float*)d_in[2];   // [256, 128]
    const float* b1 = (const float*)d_in[3];   // [128]
    const float* W2 = (const float*)d_in[4];   // [128, 16]
    const float* b2 = (const float*)d_in[5];   // [16]
    float* out = (float*)d_out;                // [N, 16]

    const int E = in_sizes[1] / 2;
    const int* src = ei;
    const int* dst = ei + E;

    // Workspace partition (256B aligned): ~58 MB total
    char* w = (char*)d_ws;
    auto take = [&](size_t bytes) -> char* {
        char* p = w;
        w += (bytes + 255) & ~(size_t)255;
        return p;
    };
    int*   cnt  = (int*)  take((size_t)N_NODES * sizeof(int));
    float* dinv = (float*)take((size_t)N_NODES * sizeof(float));
    float* h    = (float*)take((size_t)N_NODES * HIDF * sizeof(float));  // x @ W1
    float* agg1 = (float*)take((size_t)N_NODES * HIDF * sizeof(float));  // then relu(h1)
    float* h2   = (float*)take((size_t)N_NODES * NCLS * sizeof(float));  // relu_h @ W2
    float* agg2 = (float*)take((size_t)N_NODES * NCLS * sizeof(float));

    hipMemsetAsync(cnt,  0, (size_t)N_NODES * sizeof(int), stream);
    hipMemsetAsync(agg1, 0, (size_t)N_NODES * HIDF * sizeof(float), stream);
    hipMemsetAsync(agg2, 0, (size_t)N_NODES * NCLS * sizeof(float), stream);

    // 1) degrees (incl. self loop) and D^-1/2
    gcn_deg_count<<<(E + 255) / 256, 256, 0, stream>>>(dst, E, cnt);
    gcn_dinv<<<(N_NODES + 255) / 256, 256, 0, stream>>>(cnt, dinv, N_NODES);

    // 2) h = x @ W1   (WMMA fp32, 3125 x 8 tiles, 1 wave each)
    gcn_wmma_gemm_f32<F_INF, HIDF>
        <<<dim3(N_NODES / 16, HIDF / 16), 32, 0, stream>>>(x, W1, h);

    // 3) scatter over edges, then self-loop + bias + ReLU (agg1 becomes relu_h)
    gcn_agg_hid<<<(E + 7) / 8, 256, 0, stream>>>(src, dst, dinv, h, agg1, E);
    gcn_final1<<<(N_NODES * HIDF + 255) / 256, 256, 0, stream>>>(h, dinv, b1, agg1);

    // 4) h2 = relu_h @ W2  (WMMA fp32, 3125 x 1 tiles)
    gcn_wmma_gemm_f32<HIDF, NCLS>
        <<<dim3(N_NODES / 16, NCLS / 16), 32, 0, stream>>>(agg1, W2, h2);

    // 5) scatter over edges, then self-loop + bias -> d_out
    gcn_agg_out<<<(E * NCLS + 255) / 256, 256, 0, stream>>>(src, dst, dinv, h2, agg2, E);
    gcn_final2<<<(N_NODES * NCLS + 255) / 256, 256, 0, stream>>>(h2, dinv, b2, agg2, out);
}